// SimpleMemory_42391327212121
// MI455X (gfx1250) — compile-verified
//
#include <hip/hip_runtime.h>
#include <math.h>

// NCE memory-bank update + batched gather-dot for MI455X (gfx1250, wave32).
//
// Roofline: second-gather traffic ~134 MB @ 23.3 TB/s ~= 5.8 us floor; useful
// math is 67 MFLOP (0.5 FLOP/B) -> bandwidth bound. We run the dots on
// V_WMMA_F32_16X16X4_F32 (exact fp32, matches reference precision); the 16x
// broadcast-column waste (~1.07 GFLOP total) still hides under memory time.

typedef __attribute__((ext_vector_type(2))) float v2f;
typedef __attribute__((ext_vector_type(8))) float v8f;

#define FEAT 128
#define MOM  0.5f

// ---------------------------------------------------------------------------
// Kernel 1: clear the row-owner map (-1 = row not updated this call).
// ---------------------------------------------------------------------------
__global__ void nce_clear_owner(int* __restrict__ owner, int n) {
  int i = blockIdx.x * blockDim.x + threadIdx.x;
  if (i < n) owner[i] = -1;
}

// ---------------------------------------------------------------------------
// Kernel 2: one wave32 per batch row b:
//   wm[b] = normalize(MOM * bank[y[b]] + (1-MOM) * feature[b])
//   owner[y[b]] = max(owner[y[b]], b)   // last-write-wins for duplicate y
// Lane i owns elements 4i..4i+3 (float4); sum-of-squares via 5x shfl_xor.
// ---------------------------------------------------------------------------
__global__ void nce_update(const float* __restrict__ feat,
                           const float* __restrict__ bank,
                           const int*   __restrict__ y,
                           float*       __restrict__ wm,
                           int*         __restrict__ owner,
                           int B) {
  int gtid = blockIdx.x * blockDim.x + threadIdx.x;
  int b    = gtid >> 5;
  int lane = threadIdx.x & 31;
  if (b >= B) return;

  int row = y[b];
  const float4 w0 = ((const float4*)(bank + (long long)row * FEAT))[lane];
  const float4 f0 = ((const float4*)(feat + (long long)b   * FEAT))[lane];

  float4 w;
  w.x = MOM * w0.x + (1.0f - MOM) * f0.x;
  w.y = MOM * w0.y + (1.0f - MOM) * f0.y;
  w.z = MOM * w0.z + (1.0f - MOM) * f0.z;
  w.w = MOM * w0.w + (1.0f - MOM) * f0.w;

  float s = w.x * w.x + w.y * w.y + w.z * w.z + w.w * w.w;
#pragma unroll
  for (int m = 16; m >= 1; m >>= 1) s += __shfl_xor(s, m, 32);

  float inv = 1.0f / sqrtf(s);
  w.x *= inv; w.y *= inv; w.z *= inv; w.w *= inv;

  ((float4*)(wm + (long long)b * FEAT))[lane] = w;
  if (lane == 0) atomicMax(owner + row, b);
}

// ---------------------------------------------------------------------------
// Kernel 3: out[b,k] = dot(updated_bank[idx[b,k]], feature[b]) via
// V_WMMA_F32_16X16X4_F32 chained over K=128 (32 steps).
//
// A (16x4 f32): lane m<16  -> VGPR0=A[m,d+0], VGPR1=A[m,d+1]
//               lane 16+m  -> VGPR0=A[m,d+2], VGPR1=A[m,d+3]
//   -> one float2 global load per lane per step; the two lane-halves together
//      consume the full 512-B row across the d-loop (stream-once).
// B (4x16 f32): broadcast feature scalars (uniform s_loads + cndmask).
// D: columns identical; lane 0 holds k-rows 0..7, lane 16 holds 8..15.
// One block per b; 4 wave32s, one 16-k tile each.
// ---------------------------------------------------------------------------
__global__ void nce_dot_wmma(const float* __restrict__ feat,
                             const float* __restrict__ bank,
                             const int*   __restrict__ idx,
                             const float* __restrict__ wm,
                             const int*   __restrict__ owner,
                             float*       __restrict__ out,
                             int K) {
  const int  b    = blockIdx.x;
  const int  lane = threadIdx.x & 31;
  const int  wave = threadIdx.x >> 5;
  const int  nw   = blockDim.x >> 5;
  const bool hiHalf = (lane & 16) != 0;

  const float* __restrict__ f = feat + (long long)b * FEAT;  // uniform per block
  const int ktiles = K >> 4;

  for (int kt = wave; kt < ktiles; kt += nw) {
    // Each lane-pair (m, 16+m) services gathered row idx[b, kt*16 + m].
    const int kk  = (kt << 4) + (lane & 15);
    const int row = idx[(long long)b * K + kk];
    const int own = owner[row];
    const float* base = (own >= 0) ? (wm   + (long long)own * FEAT)
                                   : (bank + (long long)row * FEAT);
    const float* src = base + (hiHalf ? 2 : 0);

    v8f acc = {0.f, 0.f, 0.f, 0.f, 0.f, 0.f, 0.f, 0.f};
#pragma unroll
    for (int d = 0; d < FEAT; d += 4) {
      // Uniform scalar feature loads (SGPR), selected per lane-half.
      const float f0 = f[d + 0], f1 = f[d + 1], f2 = f[d + 2], f3 = f[d + 3];
      v2f a = *(const v2f*)(src + d);          // global_load_b64 per lane
      v2f bb;
      bb.x = hiHalf ? f2 : f0;
      bb.y = hiHalf ? f3 : f1;
      // (neg_a, A, neg_b, B, c_mod, C, reuse_a, reuse_b)
      acc = __builtin_amdgcn_wmma_f32_16x16x4_f32(
          false, a, false, bb, (short)0, acc, false, false);
    }

    // All 16 D columns are identical; lanes 0 and 16 hold the 16 k outputs.
    if ((lane & 15) == 0) {
      float* op = out + (long long)b * K + (kt << 4) + (hiHalf ? 8 : 0);
#pragma unroll
      for (int r = 0; r < 8; ++r) op[r] = acc[r];
    }
  }
}

// ---------------------------------------------------------------------------
// Inputs (setup_inputs order): feature [B*128] f32, memory_bank [N*128] f32,
// y [B] int, idx [B*K] int.  Output: [B*K] f32.
// Workspace: wm (B*128 f32, 2 MB) | owner (N int, 4 MB).
// ---------------------------------------------------------------------------
extern "C" void kernel_launch(void* const* d_in, const int* in_sizes, int n_in,
                              void* d_out, int out_size, void* d_ws, size_t ws_size,
                              hipStream_t stream) {
  const float* feat = (const float*)d_in[0];
  const float* bank = (const float*)d_in[1];
  const int*   y    = (const int*)d_in[2];
  const int*   idx  = (const int*)d_in[3];
  float*       out  = (float*)d_out;

  const int B = in_sizes[0] / FEAT;   // 4096
  const int N = in_sizes[1] / FEAT;   // 1,000,000
  const int K = in_sizes[3] / B;      // 64

  float* wm    = (float*)d_ws;
  int*   owner = (int*)((char*)d_ws + (size_t)B * FEAT * sizeof(float));

  nce_clear_owner<<<(N + 255) / 256, 256, 0, stream>>>(owner, N);
  nce_update<<<(B * 32 + 127) / 128, 128, 0, stream>>>(feat, bank, y, wm, owner, B);
  nce_dot_wmma<<<B, 128, 0, stream>>>(feat, bank, idx, wm, owner, out, K);
}